// feature_loss_39676907880503
// MI455X (gfx1250) — compile-verified
//
#include <hip/hip_runtime.h>

// Problem constants (match reference).
#define C_CLASSES 65536
#define D_DIM     512
#define B_BATCH   4096
#define DECAYF    0.9f

typedef __attribute__((ext_vector_type(2))) float v2f;
typedef __attribute__((ext_vector_type(4))) float v4f;
typedef __attribute__((ext_vector_type(8))) float v8f;

// ---------------------------------------------------------------------------
// Kernel 1: zero the per-class hit counts.
// ---------------------------------------------------------------------------
__global__ void zero_counts_kernel(int* __restrict__ counts) {
  int i = blockIdx.x * blockDim.x + threadIdx.x;
  if (i < C_CLASSES) counts[i] = 0;
}

// ---------------------------------------------------------------------------
// Kernel 2: for each update j, compute
//   ranks[j]  = #{j' > j : class[j'] == class[j]}   (suffix rank)
//   counts[c] = #{j : class[j] == c}
// O(B^2) with the class list staged in LDS; B=4096 so this is trivial.
// counts[] store is racy but every writer stores the identical value.
// ---------------------------------------------------------------------------
__global__ void rank_kernel(const int* __restrict__ cls,
                            int* __restrict__ counts,
                            int* __restrict__ ranks) {
  __shared__ int scls[B_BATCH];
  for (int i = threadIdx.x; i < B_BATCH; i += blockDim.x) scls[i] = cls[i];
  __syncthreads();

  int j = blockIdx.x * blockDim.x + threadIdx.x;
  if (j >= B_BATCH) return;
  int cj = scls[j];
  int k = 0, r = 0;
  for (int j2 = 0; j2 < B_BATCH; ++j2) {
    int c2 = scls[j2];
    if (c2 == cj) {
      ++k;
      if (j2 > j) ++r;
    }
  }
  ranks[j]  = r;
  counts[cj] = k;
}

// ---------------------------------------------------------------------------
// Kernel 3: fused EMA-compose + per-row L1 reduction via WMMA.
// One wave (32 lanes) handles 16 consecutive bank rows.
//
// Closed form per row c with k = counts[c] hits:
//   s_new[c] = 0.9^k * s[c] + sum_j 0.1 * 0.9^{ranks[j]} * feature[j]
// (commutative weighted sum -> no sequential dependence).
//
// WMMA trick: D = A(16x4 |diff|) * B(4x16 ones) + D accumulates exact f32
// row sums on the matrix pipe. Because B is all-ones, the row sum is
// invariant to which d-columns occupy which K slot, and row m is carried by
// BOTH lane m (K0/K1) and lane m+16 (K2/K3). So we partition columns:
//   lanes  0-15: row m = lane,    columns [  0..255]
//   lanes 16-31: row m = lane-16, columns [256..511]
// Each lane streams 4 contiguous floats per iteration (global_load_b128)
// and feeds two consecutive WMMAs with pairs (d0,d1) and (d2,d3).
// Every column of every row is counted exactly once across 64 iterations.
// C/D layout: lane 0 VGPR r = D[M=r, N=0]; lane 16 VGPR r = D[M=8+r, N=0].
// ---------------------------------------------------------------------------
#define LIST_CAP 1024  // matches per 16-row tile; E[matches]=1 for random data

__global__ void __launch_bounds__(32)
ema_l1_wmma_kernel(const float* __restrict__ s, const float* __restrict__ t,
                   const float* __restrict__ f, const int* __restrict__ cls,
                   const int* __restrict__ counts, const int* __restrict__ ranks,
                   float* __restrict__ out) {
  __shared__ unsigned smeta[LIST_CAP];  // (row_in_tile << 16) | update_index
  __shared__ float    sw[LIST_CAP];     // 0.1 * 0.9^rank

  const unsigned lane = threadIdx.x;        // 0..31
  const int      c0   = blockIdx.x * 16;    // first row of this tile
  const unsigned m    = lane & 15u;         // row within tile (A-frag M)
  const unsigned half = (lane >> 4) * 256u; // column half owned by this lane

  // ---- Phase 1: deterministic (ballot-ordered) compaction of hits ----------
  int nlist = 0;
  for (int j0 = 0; j0 < B_BATCH; j0 += 32) {
    int j = j0 + (int)lane;                 // B divisible by 32: always valid
    unsigned mm = (unsigned)(cls[j] - c0);
    bool match = mm < 16u;
    unsigned long long bal = __ballot(match);
    int pos = nlist + (int)__popcll(bal & ((1ull << lane) - 1ull));
    if (match && pos < LIST_CAP) {
      smeta[pos] = (mm << 16) | (unsigned)j;
      sw[pos]    = 0.1f * __powf(DECAYF, (float)ranks[j]);
    }
    nlist += (int)__popcll(bal);
  }
  int nMatch = nlist < LIST_CAP ? nlist : LIST_CAP;
  __syncthreads();

  // ---- Phase 2: stream rows (b128), compose, |diff|, 2x WMMA row-sum -------
  const float scale = __powf(DECAYF, (float)counts[c0 + (int)m]);  // k=0 -> 1.0
  const float* srow = s + (size_t)(c0 + m) * D_DIM + half;
  const float* trow = t + (size_t)(c0 + m) * D_DIM + half;

  v8f acc = {};
  v2f ones = {1.0f, 1.0f};

  for (int ch = 0; ch < 64; ++ch) {         // 64 iters x 4 cols/lane
    int off = ch * 4;
    if ((ch & 7) == 0) {  // one prefetch per 128B cacheline, ~1KB ahead
      __builtin_prefetch(srow + off + 256, 0, 0);
      __builtin_prefetch(trow + off + 256, 0, 0);
    }
    v4f sv = *(const v4f*)(srow + off);
    v4f tv = *(const v4f*)(trow + off);
    float v0 = sv.x * scale;
    float v1 = sv.y * scale;
    float v2 = sv.z * scale;
    float v3 = sv.w * scale;
    for (int i = 0; i < nMatch; ++i) {      // uniform trip count; usually 0
      unsigned meta = smeta[i];
      if ((meta >> 16) == m) {              // lanes holding the hit row
        const v4f fv =
            *(const v4f*)(f + (size_t)(meta & 0xffffu) * D_DIM + half + off);
        float w = sw[i];
        v0 += w * fv.x;
        v1 += w * fv.y;
        v2 += w * fv.z;
        v3 += w * fv.w;
      }
    }
    v2f a0, a1;
    a0.x = fabsf(v0 - tv.x);
    a0.y = fabsf(v1 - tv.y);
    a1.x = fabsf(v2 - tv.z);
    a1.y = fabsf(v3 - tv.w);
    acc = __builtin_amdgcn_wmma_f32_16x16x4_f32(
        false, a0, false, ones, (short)0, acc, false, false);
    acc = __builtin_amdgcn_wmma_f32_16x16x4_f32(
        false, a1, false, ones, (short)0, acc, false, false);
  }

  // ---- Write out: column 0 of the D tile holds all 16 row sums -------------
  if (lane == 0) {
#pragma unroll
    for (int r = 0; r < 8; ++r) out[c0 + r] = acc[r];
  } else if (lane == 16) {
#pragma unroll
    for (int r = 0; r < 8; ++r) out[c0 + 8 + r] = acc[r];
  }
}

// ---------------------------------------------------------------------------
// Host launcher.
//   d_in[0] s_feature  f32 [C*D]
//   d_in[1] t_feature  f32 [C*D]
//   d_in[2] feature    f32 [B*D]
//   d_in[3] the_class  i32 [B]
//   d_out   f32 [C]
//   d_ws: counts (C ints) | ranks (B ints)   => 272 KB used
// ---------------------------------------------------------------------------
extern "C" void kernel_launch(void* const* d_in, const int* in_sizes, int n_in,
                              void* d_out, int out_size, void* d_ws, size_t ws_size,
                              hipStream_t stream) {
  const float* s   = (const float*)d_in[0];
  const float* t   = (const float*)d_in[1];
  const float* f   = (const float*)d_in[2];
  const int*   cls = (const int*)d_in[3];
  float*       out = (float*)d_out;

  int* counts = (int*)d_ws;
  int* ranks  = counts + C_CLASSES;

  zero_counts_kernel<<<C_CLASSES / 256, 256, 0, stream>>>(counts);
  rank_kernel<<<B_BATCH / 256, 256, 0, stream>>>(cls, counts, ranks);
  ema_l1_wmma_kernel<<<C_CLASSES / 16, 32, 0, stream>>>(s, t, f, cls, counts,
                                                        ranks, out);
}